// GCNModelScatFeatureOnlyVAE_481036337853
// MI455X (gfx1250) — compile-verified
//
#include <hip/hip_runtime.h>
#include <hip/hip_bf16.h>

// ---- CDNA5 WMMA vector types (wave32) ----
typedef float v2f __attribute__((ext_vector_type(2)));
typedef float v8f __attribute__((ext_vector_type(8)));

// Problem constants (match reference)
#define NN   50000
#define EE   800000
#define H2C  64
#define H1C  128
#define DINC 256
#define BN_EPS 1e-5f

// ---------------------------------------------------------------------------
// Zero a float region (graph-capture-safe replacement for hipMemsetAsync).
// ---------------------------------------------------------------------------
__global__ void zero_f32(float* __restrict__ p, size_t n) {
  size_t i = (size_t)blockIdx.x * blockDim.x + threadIdx.x;
  size_t stride = (size_t)gridDim.x * blockDim.x;
  for (; i < n; i += stride) p[i] = 0.0f;
}

// ---------------------------------------------------------------------------
// Dense GEMM C[NrowsxNcols] = A[NrowsxK] * B[KxNcols], fp32, row-major,
// using V_WMMA_F32_16X16X4_F32. One wave32 computes one 16x16 output tile,
// stepping K by 4. Grids are sized so every wave has a full tile and EXEC
// stays all-1s (WMMA requirement). Nrows%16==0, K%4==0, Ncols%16==0.
//
// A 16x4 fp32 layout (ISA 7.12.2): lanes 0-15 -> M=lane, K=kh{0,1};
//                                  lanes 16-31 -> M=lane-16, K=kh{2,3}.
// B 4x16 fp32 layout (mirror):     lanes 0-15 -> N=lane, K=kh{0,1};
//                                  lanes 16-31 -> N=lane-16, K=kh{2,3}.
// C/D 16x16 fp32: VGPR r, lanes 0-15 -> M=r, N=lane; lanes 16-31 -> M=r+8.
// ---------------------------------------------------------------------------
__global__ void wmma_gemm_f32(const float* __restrict__ A,
                              const float* __restrict__ B,
                              float* __restrict__ C,
                              int Nrows, int K, int Ncols) {
  const int lane = threadIdx.x & 31;
  const int wave = threadIdx.x >> 5;
  const int colTiles = Ncols >> 4;
  const int rowTiles = Nrows >> 4;
  int tile = blockIdx.x * (blockDim.x >> 5) + wave;
  if (tile >= rowTiles * colTiles) return;  // whole-wave uniform (grids exact)

  const int rt = tile / colTiles;
  const int ct = tile - rt * colTiles;
  const int row0 = rt << 4;
  const int col0 = ct << 4;

  const int m  = lane & 15;         // row (A) / col (B) within tile
  const int kh = (lane >> 4) << 1;  // 0 for lanes 0-15, 2 for lanes 16-31

  const float* __restrict__ arow = A + (size_t)(row0 + m) * K + kh;
  const float* __restrict__ bcol = B + (size_t)kh * Ncols + (col0 + m);

  v8f acc = {};
  for (int k = 0; k < K; k += 4) {
    // A fragment: two consecutive K values -> one 8-byte load
    v2f a = *(const v2f*)(arow + k);
    // B fragment: two rows of B (stride Ncols apart)
    v2f b;
    b.x = bcol[(size_t)k * Ncols];
    b.y = bcol[(size_t)(k + 1) * Ncols];
    // 8-arg form: (neg_a, A, neg_b, B, c_mod, C, reuse_a, reuse_b)
    acc = __builtin_amdgcn_wmma_f32_16x16x4_f32(
        false, a, false, b, (short)0, acc, false, false);
  }

  // Store D per the 16x16 f32 C/D layout
  float* __restrict__ crow = C + (size_t)col0 + m;
  const int rbase = row0 + ((lane >> 4) << 3);
#pragma unroll
  for (int r = 0; r < 8; ++r)
    crow[(size_t)(rbase + r) * Ncols] = acc[r];
}

// ---------------------------------------------------------------------------
// Edge scatter: agg[dst[e], :] += w[e] * sup[src[e], :]
// One thread handles 4 consecutive features (float4 gather, 4 fp32 atomics).
// Consecutive threads cover consecutive features of the same edge -> the
// gather of a 512B/1KB source row is fully coalesced within a wave.
// ---------------------------------------------------------------------------
__global__ void scatter_edges(const float* __restrict__ sup,
                              float* __restrict__ agg,
                              const int* __restrict__ src,
                              const int* __restrict__ dst,
                              const float* __restrict__ w,
                              int E, int D) {
  const int chunks = D >> 2;  // float4 chunks per edge (32 or 64)
  long long idx = (long long)blockIdx.x * blockDim.x + threadIdx.x;
  long long total = (long long)E * chunks;
  if (idx >= total) return;
  int e = (int)(idx / chunks);
  int f = (int)(idx - (long long)e * chunks) << 2;

  const float wt = w[e];
  const float4 v = *(const float4*)(sup + (size_t)src[e] * D + f);
  float* __restrict__ out = agg + (size_t)dst[e] * D + f;
  atomicAdd(out + 0, v.x * wt);
  atomicAdd(out + 1, v.y * wt);
  atomicAdd(out + 2, v.z * wt);
  atomicAdd(out + 3, v.w * wt);
}

// ---------------------------------------------------------------------------
// Fused ReLU (in place) + per-column sum / sum-of-squares.
// Hierarchical: LDS bins (ds_add_f32; lanes hit distinct banks since
// consecutive elements are consecutive columns) then one global atomic per
// block per column. stats[0:D)=sum, stats[D:2D)=sumsq.
// ---------------------------------------------------------------------------
__global__ void relu_colstats(float* __restrict__ agg,
                              float* __restrict__ stats,
                              int Nrows, int D) {
  __shared__ float s_sum[DINC];
  __shared__ float s_sq[DINC];
  for (int t = threadIdx.x; t < D; t += blockDim.x) {
    s_sum[t] = 0.0f;
    s_sq[t]  = 0.0f;
  }
  __syncthreads();

  const size_t total = (size_t)Nrows * D;
  const size_t stride = (size_t)gridDim.x * blockDim.x;
  for (size_t i = (size_t)blockIdx.x * blockDim.x + threadIdx.x; i < total;
       i += stride) {
    float v = agg[i];
    v = v > 0.0f ? v : 0.0f;
    agg[i] = v;                      // ReLU in place
    int c = (int)(i & (size_t)(D - 1));  // D is a power of two (128/256)
    atomicAdd(&s_sum[c], v);
    atomicAdd(&s_sq[c], v * v);
  }
  __syncthreads();

  for (int t = threadIdx.x; t < D; t += blockDim.x) {
    atomicAdd(&stats[t], s_sum[t]);
    atomicAdd(&stats[D + t], s_sq[t]);
  }
}

// ---------------------------------------------------------------------------
// BatchNorm (training mode, biased variance) applied to the ReLU'd agg.
// ---------------------------------------------------------------------------
__global__ void bn_normalize(const float* __restrict__ agg,
                             float* __restrict__ out,
                             const float* __restrict__ gamma,
                             const float* __restrict__ beta,
                             const float* __restrict__ stats,
                             int Nrows, int D) {
  size_t total = (size_t)Nrows * D;
  size_t i = (size_t)blockIdx.x * blockDim.x + threadIdx.x;
  if (i >= total) return;
  int c = (int)(i & (size_t)(D - 1));
  const float invN = 1.0f / (float)Nrows;
  const float mean = stats[c] * invN;
  const float var  = stats[D + c] * invN - mean * mean;
  const float s = gamma[c] * rsqrtf(var + BN_EPS);
  out[i] = (agg[i] - mean) * s + beta[c];
}

// ---------------------------------------------------------------------------
extern "C" void kernel_launch(void* const* d_in, const int* in_sizes, int n_in,
                              void* d_out, int out_size, void* d_ws,
                              size_t ws_size, hipStream_t stream) {
  (void)in_sizes; (void)n_in; (void)out_size; (void)ws_size;

  const float* y   = (const float*)d_in[0];  // [N, 64]
  const int*   src = (const int*)d_in[1];    // [E]
  const int*   dst = (const int*)d_in[2];    // [E]
  const float* w   = (const float*)d_in[3];  // [E]
  const float* W1  = (const float*)d_in[4];  // [64, 128]
  const float* g1  = (const float*)d_in[5];  // [128]
  const float* b1  = (const float*)d_in[6];  // [128]
  const float* W2  = (const float*)d_in[7];  // [128, 256]
  const float* g2  = (const float*)d_in[8];  // [256]
  const float* b2  = (const float*)d_in[9];  // [256]
  float* out = (float*)d_out;                // [N, 256]

  // Workspace layout (floats), ~154 MB total
  float* ws    = (float*)d_ws;
  float* sup1  = ws;                                    // N*H1
  float* sup2  = sup1 + (size_t)NN * H1C;               // N*DIN
  float* agg1  = sup2 + (size_t)NN * DINC;              // N*H1
  float* agg2  = agg1 + (size_t)NN * H1C;               // N*DIN
  float* stats1 = agg2 + (size_t)NN * DINC;             // 2*H1
  float* stats2 = stats1 + 2 * H1C;                     // 2*DIN

  // 1) Zero accumulators + stats (agg1..stats2 are contiguous)
  const size_t zeroN =
      (size_t)NN * H1C + (size_t)NN * DINC + 2 * (H1C + DINC);
  zero_f32<<<2048, 256, 0, stream>>>(agg1, zeroN);

  // 2) support1 = y @ W1   (50000x64x128): 3125*8 = 25000 tiles, 8 waves/blk
  wmma_gemm_f32<<<25000 / 8, 256, 0, stream>>>(y, W1, sup1, NN, H2C, H1C);

  // 3) agg1 = segment_sum(w * support1[src], dst)
  {
    long long total = (long long)EE * (H1C / 4);
    int blocks = (int)((total + 255) / 256);
    scatter_edges<<<blocks, 256, 0, stream>>>(sup1, agg1, src, dst, w, EE, H1C);
  }

  // 4) ReLU in place + column stats
  relu_colstats<<<960, 256, 0, stream>>>(agg1, stats1, NN, H1C);

  // 5) h1 = BN(relu(agg1))  -> reuse sup1 as h1
  {
    size_t total = (size_t)NN * H1C;
    bn_normalize<<<(int)((total + 255) / 256), 256, 0, stream>>>(
        agg1, sup1, g1, b1, stats1, NN, H1C);
  }

  // 6) support2 = h1 @ W2   (50000x128x256): 3125*16 = 50000 tiles
  wmma_gemm_f32<<<50000 / 8, 256, 0, stream>>>(sup1, W2, sup2, NN, H1C, DINC);

  // 7) agg2 = segment_sum(w * support2[src], dst)
  {
    long long total = (long long)EE * (DINC / 4);
    int blocks = (int)((total + 255) / 256);
    scatter_edges<<<blocks, 256, 0, stream>>>(sup2, agg2, src, dst, w, EE, DINC);
  }

  // 8) ReLU in place + column stats
  relu_colstats<<<960, 256, 0, stream>>>(agg2, stats2, NN, DINC);

  // 9) out = BN(relu(agg2))
  {
    size_t total = (size_t)NN * DINC;
    bn_normalize<<<(int)((total + 255) / 256), 256, 0, stream>>>(
        agg2, out, g2, b2, stats2, NN, DINC);
  }
}